// STIDGCN_85255100825856
// MI455X (gfx1250) — compile-verified
//
#include <hip/hip_runtime.h>
#include <math.h>

// ===========================================================================
// STIDGCN forward, fp32, MI455X (gfx1250).
//
// Heavy GEMMs (adjacency builds: 1024x1024xK per batch; GCN propagation:
// 64x1024x1024 per (b,t)) run on V_WMMA_F32_16X16X4_F32 — full fp32 so the
// softmax / exact-top-k adjacency path matches the fp32 reference.
// Tiles are staged with GLOBAL_LOAD_ASYNC_TO_LDS_* (ASYNCcnt-tracked, no VGPR
// round trip) and double-buffered in LDS so tile k+1 streams in behind the
// WMMAs of tile k.  ~165 GFLOP total, adjacency working set L2-resident
// (192MB L2) -> matrix-pipe bound, not HBM bound (23.3 TB/s).
// ===========================================================================

typedef float v2f __attribute__((ext_vector_type(2)));
typedef float v8f __attribute__((ext_vector_type(8)));

#define NB 8
#define NN 1024
#define NC 64

static inline long cdiv(long a, long b) { return (a + b - 1) / b; }

// --- CDNA5 async global->LDS copy helpers (ASYNCcnt-tracked) ---------------
__device__ __forceinline__ void async_ld_b32(unsigned lds, unsigned long long ga)
{
    asm volatile("global_load_async_to_lds_b32 %0, %1, off"
                 :: "v"(lds), "v"(ga) : "memory");
}
__device__ __forceinline__ void async_ld_b128(unsigned lds, unsigned long long ga)
{
    asm volatile("global_load_async_to_lds_b128 %0, %1, off"
                 :: "v"(lds), "v"(ga) : "memory");
}
__device__ __forceinline__ void wait_async_10()
{
    asm volatile("s_wait_asynccnt 0xa" ::: "memory");
}
__device__ __forceinline__ void wait_async_0()
{
    asm volatile("s_wait_asynccnt 0x0" ::: "memory");
}

// ---------------------------------------------------------------------------
// Strided batched GEMM: C[m,n] = sum_k A[m,k]*B[k,n]
//  A: strides (rsA, csA);  B: row-major, leading dim rsB (col stride 1, base
//  and rsB*4 both 16B-aligned);  C: strides (rsC, csC).
//  batch z = zb*Tdim + zt, pointer offsets zb*bX1 + zt*bX2.
// Call-site guarantees: M,N multiples of 64; K multiple of 16.
// Block = 128 threads (4 waves). Block tile 64x64; wave w computes rows
// 16w..16w+15 x all 64 cols (4 accumulators of 16x16).
// Per 16-deep K-step each wave issues exactly 10 async ops (8x B32 for the
// strided A tile, 2x B128 for the contiguous B tile); with in-order async
// completion, s_wait_asynccnt 10 after prefetching the next tile means the
// current tile is resident.
// WMMA fragment layout per CDNA5 ISA 7.12.2 (wave32):
//  A 16x4 : lane<16 -> (K=kk,kk+1), lane>=16 -> (K=kk+2,kk+3), M=lane&15
//  B 4x16 : mirrored; C/D 16x16: vgpr v -> M=v+8*(lane>=16), N=lane&15
// ---------------------------------------------------------------------------
__global__ __launch_bounds__(128)
void gemm_f32_wmma(const float* __restrict__ Aall, const float* __restrict__ Ball,
                   float* __restrict__ Call,
                   int M, int Nn, int K,
                   long rsA, long csA, long rsB, long rsC, long csC,
                   int Tdim, long bA1, long bA2, long bB1, long bB2,
                   long bC1, long bC2)
{
    __shared__ __align__(16) float As[2][64][17];
    __shared__ __align__(16) float Bs[2][16][68];

    int zb = blockIdx.z / Tdim;
    int zt = blockIdx.z % Tdim;
    const float* A = Aall + (long)zb * bA1 + (long)zt * bA2;
    const float* Bm = Ball + (long)zb * bB1 + (long)zt * bB2;
    float* C = Call + (long)zb * bC1 + (long)zt * bC2;

    int col0 = blockIdx.x * 64;
    int row0 = blockIdx.y * 64;
    int tid  = threadIdx.x;
    int lane = tid & 31;
    int l16  = lane & 15;
    int khalf = (lane >> 4) << 1;       // 0 or 2
    int wrow = (tid >> 5) * 16;         // wave's row offset in block tile

    // issue the 10 async copies staging K-step kt into buffer b
    auto stage = [&](int kt, int b) {
        int k0 = kt << 4;
        unsigned abase = (unsigned)(unsigned long long)&As[b][0][0];
        unsigned bbase = (unsigned)(unsigned long long)&Bs[b][0][0];
#pragma unroll
        for (int e0 = 0; e0 < 8; ++e0) {               // A: 64x16, 8 elems/thread
            int e = e0 * 128 + tid;
            int i = e >> 4, kk = e & 15;
            unsigned lds = abase + (unsigned)((i * 17 + kk) << 2);
            const float* gp = A + (long)(row0 + i) * rsA + (long)(k0 + kk) * csA;
            async_ld_b32(lds, (unsigned long long)gp);
        }
#pragma unroll
        for (int e0 = 0; e0 < 2; ++e0) {               // B: 16x64, 2 x 16B/thread
            int e = e0 * 128 + tid;
            int kk = e >> 4, seg = e & 15;
            unsigned lds = bbase + (unsigned)((kk * 68 + seg * 4) << 2);
            const float* gp = Bm + (long)(k0 + kk) * rsB + (col0 + seg * 4);
            async_ld_b128(lds, (unsigned long long)gp);
        }
    };

    v8f acc[4] = {};

    int NT = K >> 4;
    stage(0, 0);
    for (int kt = 0; kt < NT; ++kt) {
        int buf = kt & 1;
        if (kt + 1 < NT) { stage(kt + 1, buf ^ 1); wait_async_10(); }
        else             { wait_async_0(); }
        __syncthreads();
#pragma unroll
        for (int kk = 0; kk < 16; kk += 4) {
            v2f a;
            a.x = As[buf][wrow + l16][kk + khalf];
            a.y = As[buf][wrow + l16][kk + khalf + 1];
#pragma unroll
            for (int j = 0; j < 4; ++j) {
                v2f bf;
                bf.x = Bs[buf][kk + khalf][j * 16 + l16];
                bf.y = Bs[buf][kk + khalf + 1][j * 16 + l16];
                acc[j] = __builtin_amdgcn_wmma_f32_16x16x4_f32(
                    false, a, false, bf, (short)0, acc[j], false, false);
            }
        }
        __syncthreads();
    }

    int mbase = row0 + wrow + ((lane >> 4) << 3);
#pragma unroll
    for (int j = 0; j < 4; ++j) {
        int n = col0 + j * 16 + l16;
#pragma unroll
        for (int v = 0; v < 8; ++v) {
            int m = mbase + v;
            C[(long)m * rsC + (long)n * csC] = acc[j][v];
        }
    }
}

// ---------------------------------------------------------------------------
// Embedding + input projection: h[b,c,n,p], c<32: in_w@x + in_b ; c>=32: embs
// ---------------------------------------------------------------------------
__global__ void k_embed(const float* __restrict__ x, const float* __restrict__ in_w,
                        const float* __restrict__ in_b, const float* __restrict__ tid_emb,
                        const float* __restrict__ diw_emb, float* __restrict__ h)
{
    long i = (long)blockIdx.x * 256 + threadIdx.x;
    if (i >= (long)NB * NC * NN * 12) return;
    int p = (int)(i % 12); long r = i / 12;
    int n = (int)(r % NN); r /= NN;
    int c = (int)(r % NC); int b = (int)(r / NC);
    const float* xv = x + ((long)(b * 12 + p) * NN + n) * 3;
    float val;
    if (c < 32) {
        val = in_b[c] + xv[0] * in_w[c * 3 + 0] + xv[1] * in_w[c * 3 + 1] + xv[2] * in_w[c * 3 + 2];
    } else {
        int cc = c - 32;
        int tid = (int)(xv[1] * 288.0f);
        int diw = (int)(xv[2] * 7.0f);
        val = tid_emb[tid * 32 + cc] + diw_emb[diw * 32 + cc];
    }
    h[i] = val;
}

// split even/odd time steps
__global__ void k_split(const float* __restrict__ src, float* __restrict__ xe,
                        float* __restrict__ xo, int Tsrc, long total)
{
    long i = (long)blockIdx.x * 256 + threadIdx.x;
    if (i >= total) return;
    int Th = Tsrc >> 1;
    int t = (int)(i % Th); long r = i / Th;
    xe[i] = src[r * Tsrc + 2 * t];
    xo[i] = src[r * Tsrc + 2 * t + 1];
}

// ---------------------------------------------------------------------------
// branch: edge-pad(3) -> conv(k=5) -> tanh -> conv(k=3) -> tanh.
// one block per (b,n); channels x time tile in LDS.
// ---------------------------------------------------------------------------
__global__ __launch_bounds__(256)
void k_branch(const float* __restrict__ xin, const float* __restrict__ w1,
              const float* __restrict__ b1, const float* __restrict__ w2,
              const float* __restrict__ b2, float* __restrict__ out, int Th)
{
    __shared__ float xp[64][13];   // Th+6 <= 12
    __shared__ float mid[64][9];   // Th+2 <= 8
    int bn = blockIdx.x;
    int n = bn & (NN - 1), b = bn >> 10;
    int Tp = Th + 6;
    for (int e = threadIdx.x; e < 64 * Tp; e += 256) {
        int c = e / Tp, j = e % Tp;
        int t = j - 3; t = t < 0 ? 0 : (t >= Th ? Th - 1 : t);
        xp[c][j] = xin[(((long)(b * 64 + c) * NN + n) * Th) + t];
    }
    __syncthreads();
    int T1 = Th + 2;
    for (int e = threadIdx.x; e < 64 * T1; e += 256) {
        int o = e / T1, tt = e % T1;
        float s = b1[o];
#pragma unroll 4
        for (int i = 0; i < 64; ++i) {
            const float* wv = w1 + (o * 64 + i) * 5;
            const float* xr = &xp[i][tt];
            s += xr[0] * wv[0] + xr[1] * wv[1] + xr[2] * wv[2] + xr[3] * wv[3] + xr[4] * wv[4];
        }
        mid[o][tt] = tanhf(s);
    }
    __syncthreads();
    for (int e = threadIdx.x; e < 64 * Th; e += 256) {
        int o = e / Th, tt = e % Th;
        float s = b2[o];
#pragma unroll 4
        for (int i = 0; i < 64; ++i) {
            const float* wv = w2 + (o * 64 + i) * 3;
            s += mid[i][tt] * wv[0] + mid[i][tt + 1] * wv[1] + mid[i][tt + 2] * wv[2];
        }
        out[(((long)(b * 64 + o) * NN + n) * Th) + tt] = tanhf(s);
    }
}

// xs[b,n,c] = sum_t z[b,c,n,t]
__global__ void k_xs(const float* __restrict__ z, float* __restrict__ xs, int Th)
{
    long i = (long)blockIdx.x * 256 + threadIdx.x;
    if (i >= (long)NB * NN * NC) return;
    int c = (int)(i & 63); long r = i >> 6;
    int n = (int)(r & (NN - 1)); int b = (int)(r >> 10);
    const float* zp = z + (((long)(b * 64 + c) * NN + n) * Th);
    float s = 0.f;
    for (int t = 0; t < Th; ++t) s += zp[t];
    xs[i] = s;
}

// A2[b,n, c*Th+t] = z[b,c,n,t]
__global__ void k_a2t(const float* __restrict__ z, float* __restrict__ A2, int Th, long total)
{
    long i = (long)blockIdx.x * 256 + threadIdx.x;
    if (i >= total) return;
    int t = (int)(i % Th); long r = i / Th;
    int c = (int)(r % 64); r /= 64;
    int n = (int)(r % NN); int b = (int)(r / NN);
    A2[i] = z[(((long)(b * 64 + c) * NN + n) * Th) + t];
}

// embT[(c*Th+t), m] = emb[c, m, t]
__global__ void k_embt(const float* __restrict__ emb, float* __restrict__ embT, int Th, long total)
{
    long i = (long)blockIdx.x * 256 + threadIdx.x;
    if (i >= total) return;
    int m = (int)(i % NN); long r = i / NN;
    int t = (int)(r % Th); int c = (int)(r / Th);
    embT[i] = emb[((long)(c * NN + m)) * Th + t];
}

__device__ __forceinline__ float blk_max(float v, float* red)
{
    int t = threadIdx.x;
    red[t] = v; __syncthreads();
    for (int s = 128; s > 0; s >>= 1) {
        if (t < s) red[t] = fmaxf(red[t], red[t + s]);
        __syncthreads();
    }
    float r = red[0]; __syncthreads();
    return r;
}
__device__ __forceinline__ float blk_sum(float v, float* red)
{
    int t = threadIdx.x;
    red[t] = v; __syncthreads();
    for (int s = 128; s > 0; s >>= 1) {
        if (t < s) red[t] += red[t + s];
        __syncthreads();
    }
    float r = red[0]; __syncthreads();
    return r;
}

// adj_row = softmax(w0*softmax(relu(a1*s)) + w1*softmax(relu(a2*s)) + fb)
// writes result in place over a1.  one block per (b,n) row of 1024.
__global__ __launch_bounds__(256)
void k_combine(float* __restrict__ a1, const float* __restrict__ a2,
               const float* __restrict__ fc_w, const float* __restrict__ fc_b)
{
    __shared__ float red[256];
    long row = blockIdx.x;
    float* p1 = a1 + row * (long)NN;
    const float* p2 = a2 + row * (long)NN;
    int t = threadIdx.x;
    float w0 = fc_w[0], w1 = fc_w[1], fb = fc_b[0];
    float v1[4], v2[4];
#pragma unroll
    for (int j = 0; j < 4; ++j) {
        float x = p1[t + 256 * j]; v1[j] = x > 0.f ? x * 0.125f : 0.f;
        float y = p2[t + 256 * j]; v2[j] = y > 0.f ? y * 0.125f : 0.f;
    }
    float m = fmaxf(fmaxf(v1[0], v1[1]), fmaxf(v1[2], v1[3]));
    m = blk_max(m, red);
    float s = 0.f;
#pragma unroll
    for (int j = 0; j < 4; ++j) { v1[j] = expf(v1[j] - m); s += v1[j]; }
    s = blk_sum(s, red);
    float inv = 1.0f / s;
#pragma unroll
    for (int j = 0; j < 4; ++j) v1[j] *= inv;

    m = fmaxf(fmaxf(v2[0], v2[1]), fmaxf(v2[2], v2[3]));
    m = blk_max(m, red);
    s = 0.f;
#pragma unroll
    for (int j = 0; j < 4; ++j) { v2[j] = expf(v2[j] - m); s += v2[j]; }
    s = blk_sum(s, red);
    inv = 1.0f / s;
#pragma unroll
    for (int j = 0; j < 4; ++j) v2[j] *= inv;

    float c[4];
#pragma unroll
    for (int j = 0; j < 4; ++j) c[j] = w0 * v1[j] + w1 * v2[j] + fb;
    m = fmaxf(fmaxf(c[0], c[1]), fmaxf(c[2], c[3]));
    m = blk_max(m, red);
    s = 0.f;
#pragma unroll
    for (int j = 0; j < 4; ++j) { c[j] = expf(c[j] - m); s += c[j]; }
    s = blk_sum(s, red);
    inv = 1.0f / s;
#pragma unroll
    for (int j = 0; j < 4; ++j) p1[t + 256 * j] = c[j] * inv;
}

// ---------------------------------------------------------------------------
// Exact top-k (k=819) per row of 1024, in place.  Radix-select on float bits
// (values are softmax outputs >= 0 -> uint order == float order); ties at the
// threshold kept lowest-index-first to match lax.top_k.
// ---------------------------------------------------------------------------
__global__ __launch_bounds__(256)
void k_topk(float* __restrict__ adj, int kkeep)
{
    __shared__ unsigned hist[256];
    __shared__ unsigned scanbuf[256];
    __shared__ unsigned sel, rrem;
    long row = blockIdx.x;
    float* p = adj + row * (long)NN;
    int t = threadIdx.x;
    unsigned bits[4];
#pragma unroll
    for (int j = 0; j < 4; ++j) bits[j] = __float_as_uint(p[t * 4 + j]);
    if (t == 0) rrem = (unsigned)kkeep;
    unsigned prefix = 0, pmask = 0;
    __syncthreads();
    for (int pass = 0; pass < 4; ++pass) {
        int shift = 24 - pass * 8;
        hist[t] = 0;
        __syncthreads();
#pragma unroll
        for (int j = 0; j < 4; ++j)
            if ((bits[j] & pmask) == prefix) atomicAdd(&hist[(bits[j] >> shift) & 255u], 1u);
        __syncthreads();
        if (t == 0) {
            unsigned r = rrem, cum = 0, bsel = 0;
            for (int bv = 255; bv >= 0; --bv) {
                unsigned hc = hist[bv];
                if (cum + hc >= r) { bsel = (unsigned)bv; rrem = r - cum; break; }
                cum += hc;
            }
            sel = bsel;
        }
        __syncthreads();
        prefix |= (sel << shift);
        pmask |= (255u << shift);
    }
    unsigned thr = prefix;
    unsigned need = rrem;
    unsigned myeq = 0;
#pragma unroll
    for (int j = 0; j < 4; ++j) if (bits[j] == thr) myeq++;
    scanbuf[t] = myeq;
    __syncthreads();
    for (int s = 1; s < 256; s <<= 1) {
        unsigned add = (t >= s) ? scanbuf[t - s] : 0u;
        __syncthreads();
        scanbuf[t] += add;
        __syncthreads();
    }
    unsigned rank = scanbuf[t] - myeq;   // exclusive prefix of equal count
#pragma unroll
    for (int j = 0; j < 4; ++j) {
        bool keep;
        if (bits[j] > thr) keep = true;
        else if (bits[j] == thr) { keep = (rank < need); rank++; }
        else keep = false;
        if (!keep) p[t * 4 + j] = 0.0f;
    }
}

// 1x1 conv over concat([h1;h2]) channels (128 -> 64)
__global__ __launch_bounds__(256)
void k_gcnconv(const float* __restrict__ h1, const float* __restrict__ h2,
               const float* __restrict__ W, const float* __restrict__ bias,
               float* __restrict__ out, int Th)
{
    __shared__ float buf[128][7];
    int bn = blockIdx.x;
    int n = bn & (NN - 1), b = bn >> 10;
    for (int e = threadIdx.x; e < 64 * Th; e += 256) {
        int c = e / Th, t = e % Th;
        buf[c][t]      = h1[(((long)(b * 64 + c) * NN + n) * Th) + t];
        buf[c + 64][t] = h2[(((long)(b * 64 + c) * NN + n) * Th) + t];
    }
    __syncthreads();
    for (int e = threadIdx.x; e < 64 * Th; e += 256) {
        int o = e / Th, t = e % Th;
        float s = bias[o];
        const float* wr = W + o * 128;
#pragma unroll 8
        for (int i = 0; i < 128; ++i) s += buf[i][t] * wr[i];
        out[(((long)(b * 64 + o) * NN + n) * Th) + t] = s;
    }
}

__global__ void k_mul_tanh(const float* __restrict__ a, const float* __restrict__ g,
                           float* __restrict__ out, long total)
{
    long i = (long)blockIdx.x * 256 + threadIdx.x;
    if (i < total) out[i] = a[i] * tanhf(g[i]);
}
__global__ void k_add(const float* __restrict__ a, const float* __restrict__ b,
                      float* __restrict__ out, long total)
{
    long i = (long)blockIdx.x * 256 + threadIdx.x;
    if (i < total) out[i] = a[i] + b[i];
}
__global__ void k_interleave(const float* __restrict__ e, const float* __restrict__ o,
                             float* __restrict__ dst, int Th, long totalE)
{
    long i = (long)blockIdx.x * 256 + threadIdx.x;
    if (i >= totalE) return;
    int t = (int)(i % Th); long r = i / Th;
    dst[r * (2 * Th) + 2 * t]     = e[i];
    dst[r * (2 * Th) + 2 * t + 1] = o[i];
}
__global__ void k_interleave_add(const float* __restrict__ i1, const float* __restrict__ i2,
                                 float* __restrict__ h, long totalE)
{
    long i = (long)blockIdx.x * 256 + threadIdx.x;
    if (i >= totalE) return;
    int t = (int)(i % 6); long r = i / 6;
    h[r * 12 + 2 * t]     += i1[i];
    h[r * 12 + 2 * t + 1] += i2[i];
}

// GLU (3x 1x1 conv + sigmoid gate) + residual + relu + output time-conv
__global__ __launch_bounds__(256)
void k_glu_out(const float* __restrict__ h,
               const float* __restrict__ w1, const float* __restrict__ b1,
               const float* __restrict__ w2, const float* __restrict__ b2,
               const float* __restrict__ w3, const float* __restrict__ b3,
               const float* __restrict__ ow, const float* __restrict__ ob,
               float* __restrict__ out)
{
    __shared__ float hb[64][13];
    __shared__ float gg[64][13];
    __shared__ float hn[64][13];
    __shared__ float red[256];
    int bn = blockIdx.x;
    int n = bn & (NN - 1), b = bn >> 10;
    for (int e = threadIdx.x; e < 64 * 12; e += 256) {
        int c = e / 12, t = e % 12;
        hb[c][t] = h[(((long)(b * 64 + c) * NN + n) * 12) + t];
    }
    __syncthreads();
    for (int e = threadIdx.x; e < 64 * 12; e += 256) {
        int o = e / 12, t = e % 12;
        float s1 = b1[o], s2 = b2[o];
#pragma unroll 8
        for (int i = 0; i < 64; ++i) {
            float x = hb[i][t];
            s1 += x * w1[o * 64 + i];
            s2 += x * w2[o * 64 + i];
        }
        gg[o][t] = s1 * (1.0f / (1.0f + expf(-s2)));
    }
    __syncthreads();
    for (int e = threadIdx.x; e < 64 * 12; e += 256) {
        int o = e / 12, t = e % 12;
        float s = b3[o];
#pragma unroll 8
        for (int i = 0; i < 64; ++i) s += gg[i][t] * w3[o * 64 + i];
        float v = s + hb[o][t];
        hn[o][t] = v > 0.f ? v : 0.f;
    }
    __syncthreads();
    int oc = threadIdx.x >> 4;
    int sub = threadIdx.x & 15;
    float part = 0.f;
    if (oc < 12) {
        for (int i = sub; i < 64; i += 16) {
            const float* wr = ow + ((long)oc * 64 + i) * 12;
#pragma unroll
            for (int k = 0; k < 12; ++k) part += hn[i][k] * wr[k];
        }
    }
    red[threadIdx.x] = part;
    __syncthreads();
    for (int s = 8; s > 0; s >>= 1) {
        if (sub < s && oc < 12) red[threadIdx.x] += red[threadIdx.x + s];
        __syncthreads();
    }
    if (sub == 0 && oc < 12)
        out[((long)(b * 12 + oc)) * NN + n] = red[threadIdx.x] + ob[oc];
}

// ===========================================================================
// Host-side orchestration
// ===========================================================================
struct Ws {
    float *h, *e1, *o1, *e2, *o2, *e3, *o3;
    float *xe, *xo, *br, *dbuf, *cbuf, *dg, *A2, *xs, *embT, *a1, *a2, *g1, *g2;
};

static void launch_gemm(hipStream_t s, const float* A, const float* B, float* C,
                        int M, int Nn, int K, long rsA, long csA, long rsB,
                        long rsC, long csC, int Tdim, long bA1, long bA2,
                        long bB1, long bB2, long bC1, long bC2, int nbz)
{
    dim3 grid((unsigned)(Nn / 64), (unsigned)(M / 64), (unsigned)nbz);
    gemm_f32_wmma<<<grid, 128, 0, s>>>(A, B, C, M, Nn, K, rsA, csA, rsB, rsC, csC,
                                       Tdim, bA1, bA2, bB1, bB2, bC1, bC2);
}

static void run_dgcn(hipStream_t s, const float* z, int Th,
                     const float* mem, const float* fcw, const float* fcb,
                     const float* gw, const float* gb, const Ws& W, float* dgout)
{
    int K2 = 64 * Th;
    k_xs<<<(unsigned)cdiv((long)NB * NN * NC, 256), 256, 0, s>>>(z, W.xs, Th);
    long totA2 = (long)NB * NN * NC * Th;
    k_a2t<<<(unsigned)cdiv(totA2, 256), 256, 0, s>>>(z, W.A2, Th, totA2);
    // a1 = xs @ mem   (per batch: 1024x1024x64)
    launch_gemm(s, W.xs, mem, W.a1, 1024, 1024, 64, 64, 1, 1024, 1024, 1,
                1, (long)NN * 64, 0, 0, 0, (long)NN * NN, 0, NB);
    // a2 = A2 @ embT  (per batch: 1024x1024xK2)
    launch_gemm(s, W.A2, W.embT, W.a2, 1024, 1024, K2, K2, 1, 1024, 1024, 1,
                1, (long)NN * K2, 0, 0, 0, (long)NN * NN, 0, NB);
    k_combine<<<NB * NN, 256, 0, s>>>(W.a1, W.a2, fcw, fcb);
    k_topk<<<NB * NN, 256, 0, s>>>(W.a1, 819);
    // h1[b,c,m,t] = sum_n z[b,c,n,t]*adj[b,n,m]   (per (b,t): 64x1024x1024)
    launch_gemm(s, z, W.a1, W.g1, 64, 1024, 1024,
                (long)NN * Th, Th, 1024, (long)NN * Th, Th,
                Th, (long)64 * NN * Th, 1, (long)NN * NN, 0, (long)64 * NN * Th, 1, NB * Th);
    launch_gemm(s, W.g1, W.a1, W.g2, 64, 1024, 1024,
                (long)NN * Th, Th, 1024, (long)NN * Th, Th,
                Th, (long)64 * NN * Th, 1, (long)NN * NN, 0, (long)64 * NN * Th, 1, NB * Th);
    k_gcnconv<<<NB * NN, 256, 0, s>>>(W.g1, W.g2, gw, gb, dgout, Th);
}

static void run_sti(hipStream_t s, const float* src, int Tsrc,
                    void* const* d_in, int base, const float* emb,
                    float* eout, float* oout, const Ws& W)
{
    auto SP = [&](int i) { return (const float*)d_in[base + i]; };
    int Th = Tsrc / 2;
    long totE = (long)NB * NC * NN * Th;
    long totT = (long)NC * NN * Th;
    k_embt<<<(unsigned)cdiv(totT, 256), 256, 0, s>>>(emb, W.embT, Th, totT);
    k_split<<<(unsigned)cdiv(totE, 256), 256, 0, s>>>(src, W.xe, W.xo, Tsrc, totE);
    const float* mem = SP(16); const float* fcw = SP(17); const float* fcb = SP(18);
    const float* gw = SP(19);  const float* gb = SP(20);
    unsigned eb = (unsigned)cdiv(totE, 256);
    // d = xo * tanh(dgcn(branch(xe, c1)))
    k_branch<<<NB * NN, 256, 0, s>>>(W.xe, SP(0), SP(1), SP(2), SP(3), W.br, Th);
    run_dgcn(s, W.br, Th, mem, fcw, fcb, gw, gb, W, W.dg);
    k_mul_tanh<<<eb, 256, 0, s>>>(W.xo, W.dg, W.dbuf, totE);
    // c = xe * tanh(dgcn(branch(xo, c2)))
    k_branch<<<NB * NN, 256, 0, s>>>(W.xo, SP(4), SP(5), SP(6), SP(7), W.br, Th);
    run_dgcn(s, W.br, Th, mem, fcw, fcb, gw, gb, W, W.dg);
    k_mul_tanh<<<eb, 256, 0, s>>>(W.xe, W.dg, W.cbuf, totE);
    // xo_u = d + dgcn(branch(c, c3))
    k_branch<<<NB * NN, 256, 0, s>>>(W.cbuf, SP(8), SP(9), SP(10), SP(11), W.br, Th);
    run_dgcn(s, W.br, Th, mem, fcw, fcb, gw, gb, W, W.dg);
    k_add<<<eb, 256, 0, s>>>(W.dbuf, W.dg, oout, totE);
    // xe_u = c + dgcn(branch(d, c4))
    k_branch<<<NB * NN, 256, 0, s>>>(W.dbuf, SP(12), SP(13), SP(14), SP(15), W.br, Th);
    run_dgcn(s, W.br, Th, mem, fcw, fcb, gw, gb, W, W.dg);
    k_add<<<eb, 256, 0, s>>>(W.cbuf, W.dg, eout, totE);
}

extern "C" void kernel_launch(void* const* d_in, const int* in_sizes, int n_in,
                              void* d_out, int out_size, void* d_ws, size_t ws_size,
                              hipStream_t stream)
{
    (void)in_sizes; (void)n_in; (void)out_size; (void)ws_size;
    // Input order: recursive insertion order of setup_inputs():
    // 0:x 1:in_w 2:in_b 3:tid_emb 4:diw_emb
    // 5..25:sti1{c1..c4{w1,b1,w2,b2}, mem, fc_w, fc_b, gcn_w, gcn_b}
    // 26..46:sti2  47..67:sti3  68:mem1 69:mem2 70:mem3
    // 71..76:glu_w1,b1,w2,b2,w3,b3  77:out_w 78:out_b
    auto P = [&](int i) { return (const float*)d_in[i]; };
    const float* X = P(0);

    float* ws = (float*)d_ws;
    Ws W;
    long off = 0;
    W.h    = ws + off; off += (long)NB * NC * NN * 12;  // 6.29M
    W.e1   = ws + off; off += (long)NB * NC * NN * 6;
    W.o1   = ws + off; off += (long)NB * NC * NN * 6;
    W.e2   = ws + off; off += (long)NB * NC * NN * 3;
    W.o2   = ws + off; off += (long)NB * NC * NN * 3;
    W.e3   = ws + off; off += (long)NB * NC * NN * 3;
    W.o3   = ws + off; off += (long)NB * NC * NN * 3;
    W.xe   = ws + off; off += (long)NB * NC * NN * 6;
    W.xo   = ws + off; off += (long)NB * NC * NN * 6;
    W.br   = ws + off; off += (long)NB * NC * NN * 6;
    W.dbuf = ws + off; off += (long)NB * NC * NN * 6;
    W.cbuf = ws + off; off += (long)NB * NC * NN * 6;
    W.dg   = ws + off; off += (long)NB * NC * NN * 6;
    W.A2   = ws + off; off += (long)NB * NC * NN * 6;
    W.xs   = ws + off; off += (long)NB * NN * NC;
    W.embT = ws + off; off += (long)NC * NN * 6;
    W.a1   = ws + off; off += (long)NB * NN * NN;
    W.a2   = ws + off; off += (long)NB * NN * NN;
    W.g1   = ws + off; off += (long)NB * NC * NN * 6;
    W.g2   = ws + off; off += (long)NB * NC * NN * 6;
    // total ~64.9M floats (~260 MB)

    long hTot = (long)NB * NC * NN * 12;
    k_embed<<<(unsigned)cdiv(hTot, 256), 256, 0, stream>>>(X, P(1), P(2), P(3), P(4), W.h);

    run_sti(stream, W.h,  12, d_in, 5,  P(68), W.e1, W.o1, W);
    run_sti(stream, W.e1,  6, d_in, 26, P(69), W.e2, W.o2, W);
    run_sti(stream, W.o1,  6, d_in, 47, P(70), W.e3, W.o3, W);

    long tot3 = (long)NB * NC * NN * 3;
    long tot6 = (long)NB * NC * NN * 6;
    k_interleave<<<(unsigned)cdiv(tot3, 256), 256, 0, stream>>>(W.e2, W.o2, W.xe, 3, tot3);
    k_interleave<<<(unsigned)cdiv(tot3, 256), 256, 0, stream>>>(W.e3, W.o3, W.xo, 3, tot3);
    k_interleave_add<<<(unsigned)cdiv(tot6, 256), 256, 0, stream>>>(W.xe, W.xo, W.h, tot6);

    k_glu_out<<<NB * NN, 256, 0, stream>>>(W.h, P(71), P(72), P(73), P(74), P(75), P(76),
                                           P(77), P(78), (float*)d_out);
}